// classic_slab1D_eqx_only_36618891166424
// MI455X (gfx1250) — compile-verified
//
#include <hip/hip_runtime.h>

// CDNA5 (gfx1250) wave32 WMMA types
typedef __attribute__((ext_vector_type(2))) float v2f;
typedef __attribute__((ext_vector_type(8))) float v8f;

#define CPW 8  // chunks of 128 timesteps handled per wave (8 WMMAs/wave)

// Complex double power by square-and-multiply: (rr + i*ri) = (br + i*bi)^e.
// <=32 short iterations for e < 2^32; far cheaper and more accurate than
// the polar exp/sincos form for phases of hundreds of radians.
__device__ __forceinline__ void cpow_u32(double br, double bi, unsigned e,
                                         double& rr, double& ri) {
  rr = 1.0;
  ri = 0.0;
  while (e) {
    if (e & 1u) {
      double t = rr * br - ri * bi;
      ri = rr * bi + ri * br;
      rr = t;
    }
    double t = br * br - bi * bi;
    bi = 2.0 * br * bi;
    br = t;
    e >>= 1u;
  }
}

// q(T) = d * S_T = g * (c^T - 1) with g = d/(c-1);  q(T) = d*T when c == 1.
__device__ __forceinline__ void qval(double pr, double pi, bool degen,
                                     double gr, double gi, double dr, double di,
                                     double T, double& qr, double& qi) {
  if (degen) {
    qr = dr * T;
    qi = di * T;
  } else {
    double xr = pr - 1.0, xi = pi;
    qr = gr * xr - gi * xi;
    qi = gr * xi + gi * xr;
  }
}

// Affine-scan-as-GEMM:  Out(16x16) = W(16x4) @ V(4x16) per 128-step chunk:
//   W row j        = [Re c^j, Im c^j, Re(d*S_j), Im(d*S_j)]     (chunk-invariant)
//   V cols 2m,2m+1 = [[x_m, y_m], [-y_m, x_m], [1, 0], [0, 1]]  (block base U)
//   Out[j][2m] = Re U[T+16m+j], Out[j][2m+1] = Im U[T+16m+j]
// Block bases advance across chunks with U[T+128] = c^128*U[T] + d*S_128.
__global__ __launch_bounds__(32) void classic_slab1D_scan_wmma(
    const float* __restrict__ pk, const float* __restrict__ taxp,
    const float* __restrict__ tayp, const float* __restrict__ fcp,
    const int* __restrict__ dtp, float* __restrict__ out, int nt) {
  const int lane = threadIdx.x & 31;
  const int j    = lane & 15;        // A row (time offset) == B column index
  const bool hi  = (lane >= 16);     // K half: K=0,1 (lo) vs K=2,3 (hi)
  const bool odd = (j & 1);
  const int m    = j >> 1;           // block index within chunk

  // Constants in fp32, exactly as the reference rounds them (K=exp(pk), c, d
  // are all complex64/f32 in the reference), then promoted to double.
  const float dtf = (float)dtp[0];
  const float K0f = expf(pk[0]);
  const float K1f = expf(pk[1]);
  const float crf = 1.0f - dtf * K1f;        // c = 1 + dt*(-i*fc - K1)
  const float cif = -dtf * fcp[0];
  const float drf = dtf * K0f * taxp[0];     // d = dt*K0*(TAx + i*TAy)
  const float dif = dtf * K0f * tayp[0];

  const double cr = crf, ci = cif, dr = drf, di = dif;
  const double er = cr - 1.0, ei = ci;
  const double den = er * er + ei * ei;
  const bool degen = (den < 1e-300);         // c == 1 -> U[t] = d*t
  double gr = 0.0, gi = 0.0;                 // g = d/(c-1)
  if (!degen) {
    gr = (dr * er + di * ei) / den;
    gi = (di * er - dr * ei) / den;
  }

  // ---- A matrix (16x4 f32), chunk-invariant ----
  double pr, pi, qr, qi;
  cpow_u32(cr, ci, (unsigned)j, pr, pi);
  qval(pr, pi, degen, gr, gi, dr, di, (double)j, qr, qi);
  v2f a;
  a.x = hi ? (float)qr : (float)pr;   // K=2: Re(d*S_j) | K=0: Re c^j
  a.y = hi ? (float)qi : (float)pi;   // K=3: Im(d*S_j) | K=1: Im c^j

  // ---- chunk-step constants: c^128 and q128 = d*S_128 ----
  double c128r, c128i, q128r, q128i;
  cpow_u32(cr, ci, 128u, c128r, c128i);
  qval(c128r, c128i, degen, gr, gi, dr, di, 128.0, q128r, q128i);

  // ---- initial block base: U[T0 + 16m] = q(T0 + 16m) ----
  const long long T0 = (long long)blockIdx.x * (128LL * CPW);
  const unsigned e0 = (unsigned)(T0 + 16 * m);
  double tr, ti, ur, ui;
  cpow_u32(cr, ci, e0, tr, ti);
  qval(tr, ti, degen, gr, gi, dr, di, (double)e0, ur, ui);

#pragma unroll
  for (int k = 0; k < CPW; ++k) {
    const int base = (int)T0 + k * 128;

    // B matrix (4x16 f32): column j, K pair per wave half
    const float x = (float)ur, y = (float)ui;
    v2f b;
    if (!hi) {                       // rows K=0,1 of V
      b.x = odd ? y : x;
      b.y = odd ? x : -y;
    } else {                         // rows K=2,3 of V (affine offset selector)
      b.x = odd ? 0.0f : 1.0f;
      b.y = odd ? 1.0f : 0.0f;
    }

    // 128 timesteps in one CDNA5 matrix op
    v8f acc = {};
    acc = __builtin_amdgcn_wmma_f32_16x16x4_f32(
        /*neg_a=*/false, a, /*neg_b=*/false, b,
        /*c_mod=*/(short)0, acc, /*reuse_a=*/false, /*reuse_b=*/false);

    // Store: lane owns 8 consecutive timesteps of one component.
    // D layout: VGPR v -> row M = v + 8*hi, column N = lane&15.
    const int tb  = base + 16 * m + (hi ? 8 : 0);
    const int off = odd ? nt : 0;    // odd columns = imaginary part
    if (((nt & 3) == 0) && (tb + 8 <= nt)) {
      *reinterpret_cast<float4*>(out + off + tb) =
          make_float4(acc[0], acc[1], acc[2], acc[3]);
      *reinterpret_cast<float4*>(out + off + tb + 4) =
          make_float4(acc[4], acc[5], acc[6], acc[7]);
    } else {
#pragma unroll
      for (int v = 0; v < 8; ++v) {
        int t = tb + v;
        if (t < nt) out[off + t] = acc[v];
      }
    }

    // advance block base by 128 steps: u = c^128 * u + d*S_128
    const double nur = c128r * ur - c128i * ui + q128r;
    const double nui = c128r * ui + c128i * ur + q128i;
    ur = nur;
    ui = nui;
  }
}

extern "C" void kernel_launch(void* const* d_in, const int* in_sizes, int n_in,
                              void* d_out, int out_size, void* d_ws, size_t ws_size,
                              hipStream_t stream) {
  (void)in_sizes; (void)n_in; (void)d_ws; (void)ws_size;
  const float* pk  = (const float*)d_in[0];  // 2 elements
  const float* tax = (const float*)d_in[1];
  const float* tay = (const float*)d_in[2];
  const float* fc  = (const float*)d_in[3];
  // d_in[4] = nt (int, recoverable from out_size), d_in[5] = dt (int)
  const int*   dt  = (const int*)d_in[5];
  float* out = (float*)d_out;

  const int nt = out_size / 2;               // output = [Re U[0:nt), Im U[0:nt)]
  if (nt <= 0) return;
  const int chunks = (nt + 127) / 128;       // 128 timesteps per WMMA
  const int blocks = (chunks + CPW - 1) / CPW;

  hipLaunchKernelGGL(classic_slab1D_scan_wmma, dim3(blocks), dim3(32), 0, stream,
                     pk, tax, tay, fc, dt, out, nt);
}